// StationInfluence_59854664237700
// MI455X (gfx1250) — compile-verified
//
#include <hip/hip_runtime.h>
#include <cstdint>

#define D_FEAT 256
#define BATCH  32
#define WAVES_PER_BLOCK 8
#define THREADS (WAVES_PER_BLOCK * 32)

// ---------------------------------------------------------------------------
// gfx1250 async Global->LDS path (ASYNCcnt-tracked), guarded so the file
// always compiles. This toolchain declares the builtin with generic `int*`
// parameters (probe round 2 diagnostic); codegen inserts the p1/p3 casts.
// ---------------------------------------------------------------------------
#if __has_builtin(__builtin_amdgcn_global_load_async_to_lds_b32)
#define HAVE_ASYNC_LDS 1
#else
#define HAVE_ASYNC_LDS 0
#endif

#if HAVE_ASYNC_LDS
#if __has_builtin(__builtin_amdgcn_s_wait_asynccnt)
#define WAIT_ASYNC(n) do { __builtin_amdgcn_s_wait_asynccnt(n); \
                           asm volatile("" ::: "memory"); } while (0)
#else
#define WAIT_ASYNC(n) asm volatile("s_wait_asynccnt %0" :: "n"(n) : "memory")
#endif
#else
#define WAIT_ASYNC(n) do {} while (0)
#endif

// ---------------------------------------------------------------------------
// Zero-fill the output (harness poisons d_out with 0xAA before timing).
// ---------------------------------------------------------------------------
__global__ void __launch_bounds__(256)
zero_out_kernel(float4* __restrict__ out4, int n4) {
    int i = blockIdx.x * blockDim.x + threadIdx.x;
    const int stride = gridDim.x * blockDim.x;
    const float4 z = make_float4(0.f, 0.f, 0.f, 0.f);
    for (; i < n4; i += stride) out4[i] = z;
}

// ---------------------------------------------------------------------------
// COO SpMM, batch-of-32-edges-per-wave pipeline:
//   stage(b+1): 3 async b32 loads per lane -> LDS double buffer (ASYNCcnt)
//   wait asynccnt<=3, ds_load this batch's (val,row,col)
//   prefetch phase: 8 x { ds_bpermute col broadcast + global_prefetch_b8 }
//                   -> warms all 8 x 128B lines of all 32 source rows in L2
//   inner loop over 32 edges: v_readlane -> SGPR metadata,
//                   2 x global_load_b128 gather of the 1KB source row,
//                   8 non-returning global_atomic_add_f32 into out[r,:]
// out (51.2 MB) and x (51.2 MB) both fit in the 192 MB L2, so gathers and
// atomics resolve at L2 bandwidth; HBM sees only ~122 MB of streaming.
// ---------------------------------------------------------------------------
__global__ void __launch_bounds__(THREADS)
spmm_coo_kernel(const float* __restrict__ x,     // [N_NODES, 256]
                const float* __restrict__ vals,  // [NNZ]
                const int*   __restrict__ rows,  // [NNZ]
                const int*   __restrict__ cols,  // [NNZ]
                float*       __restrict__ out,   // [N_NODES, 256]
                int nnz) {
    __shared__ float s_vals[2][WAVES_PER_BLOCK][BATCH];
    __shared__ int   s_rows[2][WAVES_PER_BLOCK][BATCH];
    __shared__ int   s_cols[2][WAVES_PER_BLOCK][BATCH];

    const int lane   = threadIdx.x & 31;
    const int w      = threadIdx.x >> 5;
    const int wave   = (blockIdx.x * blockDim.x + threadIdx.x) >> 5;
    const int nwaves = (gridDim.x * blockDim.x) >> 5;

    const long long bstride = (long long)nwaves * BATCH;
    const long long e0      = (long long)wave * BATCH;
    if (e0 >= nnz) return;

    // Stage one 32-edge batch of metadata into LDS buffer `buf`.
    auto stage = [&](long long base, int buf) {
        long long e = base + lane;
        if (e > (long long)nnz - 1) e = nnz - 1;  // clamp (tail lanes)
#if HAVE_ASYNC_LDS
        __builtin_amdgcn_global_load_async_to_lds_b32(
            (int*)(vals + e), (int*)&s_vals[buf][w][lane], 0, 0);
        __builtin_amdgcn_global_load_async_to_lds_b32(
            (int*)(rows + e), (int*)&s_rows[buf][w][lane], 0, 0);
        __builtin_amdgcn_global_load_async_to_lds_b32(
            (int*)(cols + e), (int*)&s_cols[buf][w][lane], 0, 0);
#else
        s_vals[buf][w][lane] = vals[e];
        s_rows[buf][w][lane] = rows[e];
        s_cols[buf][w][lane] = cols[e];
#endif
    };

    stage(e0, 0);
    int buf = 0;

    for (long long b = e0; b < nnz; b += bstride) {
        const long long bn = b + bstride;
        if (bn < nnz) {
            stage(bn, buf ^ 1);   // enqueue next batch (3 more async ops)
            WAIT_ASYNC(3);        // oldest 3 (current batch) have landed
        } else {
            WAIT_ASYNC(0);
        }

        // This wave's private LDS slice: no barrier needed.
        const float myv = s_vals[buf][w][lane];
        const int   myr = s_rows[buf][w][lane];
        const int   myc = s_cols[buf][w][lane];

        // Warm L2 with every 128B line of every source row in this batch:
        // round p covers rows p*4 .. p*4+3, lanes split 8 chunks per row.
        #pragma unroll
        for (int p = 0; p < 8; ++p) {
            const int src_lane = p * 4 + (lane >> 3);
            const int cc       = __shfl(myc, src_lane, 32);   // ds_bpermute_b32
            __builtin_prefetch(x + (size_t)cc * D_FEAT + (lane & 7) * 32,
                               /*rw=*/0, /*loc=*/1);          // global_prefetch_b8
        }

        const long long rem   = (long long)nnz - b;
        const int       limit = rem < BATCH ? (int)rem : BATCH;

        for (int i = 0; i < limit; ++i) {
            // Uniform loop index -> v_readlane_b32 puts metadata in SGPRs.
            const int   r = __builtin_amdgcn_readlane(myr, i);
            const int   c = __builtin_amdgcn_readlane(myc, i);
            const float v = __int_as_float(
                __builtin_amdgcn_readlane(__float_as_int(myv), i));

            const float4* __restrict__ src =
                (const float4*)(x + (size_t)c * D_FEAT);
            const float4 a  = src[lane];        // floats [  0..127]
            const float4 b4 = src[lane + 32];   // floats [128..255]

            float* dst = out + (size_t)r * D_FEAT + lane * 4;
            (void)__hip_atomic_fetch_add(dst + 0,   v * a.x,  __ATOMIC_RELAXED, __HIP_MEMORY_SCOPE_AGENT);
            (void)__hip_atomic_fetch_add(dst + 1,   v * a.y,  __ATOMIC_RELAXED, __HIP_MEMORY_SCOPE_AGENT);
            (void)__hip_atomic_fetch_add(dst + 2,   v * a.z,  __ATOMIC_RELAXED, __HIP_MEMORY_SCOPE_AGENT);
            (void)__hip_atomic_fetch_add(dst + 3,   v * a.w,  __ATOMIC_RELAXED, __HIP_MEMORY_SCOPE_AGENT);
            (void)__hip_atomic_fetch_add(dst + 128, v * b4.x, __ATOMIC_RELAXED, __HIP_MEMORY_SCOPE_AGENT);
            (void)__hip_atomic_fetch_add(dst + 129, v * b4.y, __ATOMIC_RELAXED, __HIP_MEMORY_SCOPE_AGENT);
            (void)__hip_atomic_fetch_add(dst + 130, v * b4.z, __ATOMIC_RELAXED, __HIP_MEMORY_SCOPE_AGENT);
            (void)__hip_atomic_fetch_add(dst + 131, v * b4.w, __ATOMIC_RELAXED, __HIP_MEMORY_SCOPE_AGENT);
        }
        buf ^= 1;
    }
}

// ---------------------------------------------------------------------------
// Inputs (setup_inputs order):
//   d_in[0] spot_x  float32 [50000*256]
//   d_in[1] A_vals  float32 [NNZ]
//   d_in[2] A_rows  int32   [NNZ]
//   d_in[3] A_cols  int32   [NNZ]
// Output: float32 [50000*256]
// ---------------------------------------------------------------------------
extern "C" void kernel_launch(void* const* d_in, const int* in_sizes, int n_in,
                              void* d_out, int out_size, void* d_ws, size_t ws_size,
                              hipStream_t stream) {
    (void)n_in; (void)d_ws; (void)ws_size;

    const float* x    = (const float*)d_in[0];
    const float* vals = (const float*)d_in[1];
    const int*   rows = (const int*)d_in[2];
    const int*   cols = (const int*)d_in[3];
    float*       out  = (float*)d_out;
    const int    nnz  = in_sizes[1];

    // 1) Zero the (poisoned) output buffer.
    const int n4 = out_size / 4;
    int zblocks = (n4 + 255) / 256;
    if (zblocks > 4096) zblocks = 4096;
    zero_out_kernel<<<zblocks, 256, 0, stream>>>((float4*)out, n4);

    // 2) Edge scatter: 2048 blocks x 8 waves = 16384 waves; 1.6M edges /
    //    (16384 * 32-edge batches) ~= 3 batches per wave, so the async
    //    double-buffer pipeline is actually exercised while keeping
    //    good occupancy for latency hiding.
    long long nbatches = ((long long)nnz + BATCH - 1) / BATCH;
    long long wantblocks = (nbatches + WAVES_PER_BLOCK - 1) / WAVES_PER_BLOCK;
    int sblocks = wantblocks > 2048 ? 2048 : (int)wantblocks;
    if (sblocks < 1) sblocks = 1;
    spmm_coo_kernel<<<sblocks, THREADS, 0, stream>>>(x, vals, rows, cols, out, nnz);
}